// ResidualNetwork_24404004176275
// MI455X (gfx1250) — compile-verified
//
#include <hip/hip_runtime.h>

typedef __attribute__((ext_vector_type(2))) float v2f;
typedef __attribute__((ext_vector_type(8))) float v8f;

#define H_DIM 10
#define L_BLOCKS 10

// LDS layout (in floats):
//   [0,64)           w0 padded to [4][16]
//   [64, 64+33*192)  33 weight matrices padded to [12][16]:
//                    slot0=w1, slots1..30=(Wr1,Wr2,Wr3) per block, 31=w8, 32=w9
//   biases: 36 vectors of [16]: b0,b1,(Br1,Br2,Br3)*10,b8,b9
//   scratch: 8 waves * 2 tiles * 256 floats
#define W0_OFF   0
#define WM_OFF   64
#define B_OFF    (64 + 33*192)          /* 6400 */
#define SCR_OFF  (B_OFF + 36*16)        /* 6976 */
#define LDS_FLOATS (SCR_OFF + 8*2*256)  /* 11072 floats = 44.3 KB */

__device__ __forceinline__ v8f wmma4(v2f a, v2f b, v8f c) {
  // D = A(16x4,f32) * B(4x16,f32) + C(16x16,f32), full fp32 precision
  return __builtin_amdgcn_wmma_f32_16x16x4_f32(
      /*neg_a=*/false, a, /*neg_b=*/false, b,
      /*c_mod=*/(short)0, c, /*reuse_a=*/false, /*reuse_b=*/false);
}

__global__ __launch_bounds__(256) void eikonal_resmlp_wmma(
    const float* __restrict__ x,
    const float* __restrict__ w0,  const float* __restrict__ b0,
    const float* __restrict__ w1,  const float* __restrict__ b1,
    const float* __restrict__ Wr1, const float* __restrict__ Br1,
    const float* __restrict__ Wr2, const float* __restrict__ Br2,
    const float* __restrict__ Wr3, const float* __restrict__ Br3,
    const float* __restrict__ w8,  const float* __restrict__ b8,
    const float* __restrict__ w9,  const float* __restrict__ b9,
    float* __restrict__ out, int npts)
{
  __shared__ float lds[LDS_FLOATS];
  const int tid = threadIdx.x;

  // ---------- cooperative zero-padded weight preload ----------
  for (int i = tid; i < 64; i += 256) {          // w0: [3][10] -> [4][16]
    int k = i >> 4, n = i & 15;
    lds[W0_OFF + i] = (k < 3 && n < H_DIM) ? w0[k * H_DIM + n] : 0.f;
  }
  for (int i = tid; i < 33 * 192; i += 256) {    // 33 mats: [10][nout] -> [12][16]
    int slot = i / 192, r = i % 192;
    int k = r >> 4, n = r & 15;
    const float* src; int nout = H_DIM;
    if (slot == 0)       src = w1;
    else if (slot <= 30) {
      int l = (slot - 1) / 3, s = (slot - 1) % 3;
      src = (s == 0 ? Wr1 : (s == 1 ? Wr2 : Wr3)) + l * H_DIM * H_DIM;
    }
    else if (slot == 31) src = w8;
    else               { src = w9; nout = 1; }
    lds[WM_OFF + i] = (k < H_DIM && n < nout) ? src[k * nout + n] : 0.f;
  }
  for (int i = tid; i < 36 * 16; i += 256) {     // biases -> [16]
    int slot = i >> 4, n = i & 15;
    const float* src; int dim = H_DIM;
    if (slot == 0)       src = b0;
    else if (slot == 1)  src = b1;
    else if (slot <= 33) {
      int l = (slot - 2) / 3, s = (slot - 2) % 3;
      src = (s == 0 ? Br1 : (s == 1 ? Br2 : Br3)) + l * H_DIM;
    }
    else if (slot == 34) src = b8;
    else               { src = b9; dim = 1; }
    lds[B_OFF + i] = (n < dim) ? src[n] : 0.f;
  }
  __syncthreads();

  const int lane = tid & 31;
  const int wave = tid >> 5;
  const int m    = lane & 15;          // A: point row / B,C,D: column
  const int kh   = (lane >> 4) << 1;   // K sub-offset within a 4-chunk: 0 or 2
  const int rowh = (lane >> 4) << 3;   // D row offset: 0 or 8

  float* Hbuf  = &lds[SCR_OFF + wave * 512];
  float* H1buf = Hbuf + 256;

  const int tile = blockIdx.x * 8 + wave;   // 16 points per wave
  const int p0   = tile * 16;
  if (p0 >= npts) return;

  auto biasC = [&](int slot) -> v8f {
    float bv = lds[B_OFF + slot * 16 + m];
    v8f c;
#pragma unroll
    for (int i = 0; i < 8; ++i) c[i] = bv;
    return c;
  };
  auto relu8 = [&](v8f c) -> v8f {
    // select form folds to a single v_max_num_f32 per element
    // (fmaxf emits canonicalize + max = 2 VALU each)
#pragma unroll
    for (int i = 0; i < 8; ++i) c[i] = (c[i] > 0.f) ? c[i] : 0.f;
    return c;
  };
  auto storeD = [&](float* buf, v8f c) {       // D-layout -> [point][feat] tile
#pragma unroll
    for (int r = 0; r < 8; ++r) buf[(r + rowh) * 16 + m] = c[r];
  };
  auto loadA = [&](const float* buf, v2f* a) { // [point][feat] -> A operands, K=12
#pragma unroll
    for (int j = 0; j < 3; ++j) {
      int kb = 4 * j + kh;
      a[j].x = buf[m * 16 + kb];
      a[j].y = buf[m * 16 + kb + 1];
    }
  };
  auto gemm = [&](const v2f* a, int wslot, v8f c) -> v8f {  // K=12 in 3 WMMA steps
    const float* W = &lds[WM_OFF + wslot * 192];
#pragma unroll
    for (int j = 0; j < 3; ++j) {
      int kb = 4 * j + kh;
      v2f b;
      b.x = W[kb * 16 + m];
      b.y = W[(kb + 1) * 16 + m];
      c = wmma4(a[j], b, c);
    }
    return c;
  };

  // ---------- layer 0: h0 = relu(x @ w0 + b0), K=3 padded to 4 ----------
  {
    int p = p0 + m;
    v2f a;
    a.x = x[p * 3 + kh];
    a.y = (kh == 0) ? x[p * 3 + 1] : 0.f;   // K=3 -> lane-half 1 second elem is pad
    v2f b;
    b.x = lds[W0_OFF + kh * 16 + m];
    b.y = lds[W0_OFF + (kh + 1) * 16 + m];
    v8f acc = biasC(0);
    acc = wmma4(a, b, acc);
    storeD(Hbuf, relu8(acc));
  }

  v2f ah[3], ah1[3];

  // ---------- layer 1: h = relu(h0 @ w1 + b1) ----------
  loadA(Hbuf, ah);
  {
    v8f acc = gemm(ah, 0, biasC(1));
    storeD(Hbuf, relu8(acc));
  }

  // ---------- residual blocks ----------
  for (int l = 0; l < L_BLOCKS; ++l) {
    loadA(Hbuf, ah);                                      // A(h), reused twice
    v8f acc = gemm(ah, 1 + 3 * l + 0, biasC(2 + 3 * l + 0));  // h @ W1 + B1
    acc = relu8(acc);                                     // h1
    storeD(H1buf, acc);
    loadA(H1buf, ah1);
    v8f acc2;
    {
      float bv = lds[B_OFF + (2 + 3 * l + 1) * 16 + m]
               + lds[B_OFF + (2 + 3 * l + 2) * 16 + m];   // B2 + B3
#pragma unroll
      for (int i = 0; i < 8; ++i) acc2[i] = bv;
    }
    acc2 = gemm(ah1, 1 + 3 * l + 2, acc2);                // + h1 @ W3
    acc2 = gemm(ah,  1 + 3 * l + 1, acc2);                // + h  @ W2
    storeD(Hbuf, relu8(acc2));
  }

  // ---------- penultimate: relu(h @ w8 + b8) ----------
  loadA(Hbuf, ah);
  {
    v8f acc = gemm(ah, 31, biasC(34));
    storeD(Hbuf, relu8(acc));
  }
  // ---------- output: h @ w9 + b9 (column 0 only) ----------
  loadA(Hbuf, ah);
  {
    v8f acc = gemm(ah, 32, biasC(35));
    if (m == 0) {                         // lanes 0 and 16 hold the N=0 column
#pragma unroll
      for (int r = 0; r < 8; ++r) out[p0 + rowh + r] = acc[r];
    }
  }
}

extern "C" void kernel_launch(void* const* d_in, const int* in_sizes, int n_in,
                              void* d_out, int out_size, void* d_ws, size_t ws_size,
                              hipStream_t stream) {
  const float* x   = (const float*)d_in[0];
  const float* w0  = (const float*)d_in[1];
  const float* b0  = (const float*)d_in[2];
  const float* w1  = (const float*)d_in[3];
  const float* b1  = (const float*)d_in[4];
  const float* Wr1 = (const float*)d_in[5];
  const float* Br1 = (const float*)d_in[6];
  const float* Wr2 = (const float*)d_in[7];
  const float* Br2 = (const float*)d_in[8];
  const float* Wr3 = (const float*)d_in[9];
  const float* Br3 = (const float*)d_in[10];
  const float* w8  = (const float*)d_in[11];
  const float* b8  = (const float*)d_in[12];
  const float* w9  = (const float*)d_in[13];
  const float* b9  = (const float*)d_in[14];
  float* out = (float*)d_out;

  const int npts = in_sizes[0] / 3;            // x is [N,3]
  const int blocks = (npts + 127) / 128;       // 8 waves/block * 16 pts/wave
  eikonal_resmlp_wmma<<<blocks, 256, 0, stream>>>(
      x, w0, b0, w1, b1, Wr1, Br1, Wr2, Br2, Wr3, Br3, w8, b8, w9, b9,
      out, npts);
}